// Floyd_61924838474225
// MI455X (gfx1250) — compile-verified
//
#include <hip/hip_runtime.h>
#include <hip/hip_bf16.h>
#include <math.h>

typedef __attribute__((ext_vector_type(2))) float v2f;
typedef __attribute__((ext_vector_type(8))) float v8f;

#define M_G      16
#define N_N      20
#define NPAIR    256          // M_G * M_G
#define NN2      400          // N_N * N_N
#define KSTRIDE  160000       // 400 * 400 floats per pair
#define NTHREADS 512          // 16 wave32s
#define CONST_W  0.3f

// Per-wave: compute aff[pair] = sum_{a1,a2} K2[pair][p(a1), p(a2)] for the 16
// pairs {wave*16 .. wave*16+15}, where p(a) = sig[pair][a]*20 + a.
// Reduction is done on the matrix pipe: A(16x4) holds 4 gathered terms per
// pair (M = pair), B = ones, D = A*B + C accumulates row sums into every
// column; after 100 WMMAs D[m][0] is the 400-term sum for pair m.
__device__ __forceinline__ void wave_aff(const float* __restrict__ K,
                                         const short* __restrict__ sig, // [256][20] in LDS
                                         float* __restrict__ dst,       // [256] in LDS
                                         int wave, int lane)
{
    const int prow  = lane & 15;
    const int pairB = wave * 16 + prow;
    const float* Kb = K + (size_t)pairB * KSTRIDE;
    const short* sp = sig + pairB * N_N;

    v8f acc = {0.f, 0.f, 0.f, 0.f, 0.f, 0.f, 0.f, 0.f};
    v2f bons; bons.x = 1.0f; bons.y = 1.0f;   // B[k][n] = 1 everywhere

    // A-matrix 16x4 layout: lanes 0-15 -> K=0 (vgpr0), K=1 (vgpr1);
    //                       lanes 16-31 -> K=2 (vgpr0), K=3 (vgpr1).
    const int tbase = (lane < 16) ? 0 : 2;
    for (int it = 0; it < 100; ++it) {
        const int t0 = it * 4 + tbase;
        const int t1 = t0 + 1;
        const int a1 = t0 / N_N, a2 = t0 % N_N;
        const int b1 = t1 / N_N, b2 = t1 % N_N;
        const int p1 = (int)sp[a1] * N_N + a1;
        const int p2 = (int)sp[a2] * N_N + a2;
        const int q1 = (int)sp[b1] * N_N + b1;
        const int q2 = (int)sp[b2] * N_N + b2;
        v2f av;
        av.x = Kb[p1 * NN2 + p2];
        av.y = Kb[q1 * NN2 + q2];
        acc = __builtin_amdgcn_wmma_f32_16x16x4_f32(
                  false, av, false, bons, (short)0, acc, false, false);
    }

    // D layout: lane 0 vgpr r = D[r][0] (rows 0-7), lane 16 vgpr r = D[8+r][0].
    if (lane == 0) {
#pragma unroll
        for (int mm = 0; mm < 8; ++mm) dst[wave * 16 + mm] = acc[mm];
    } else if (lane == 16) {
#pragma unroll
        for (int mm = 0; mm < 8; ++mm) dst[wave * 16 + 8 + mm] = acc[mm];
    }
}

__launch_bounds__(NTHREADS)
__global__ void floyd_persistent_kernel(const float* __restrict__ K,
                                        const float* __restrict__ Xin,
                                        float* __restrict__ out)
{
    __shared__ short sigma[NPAIR][N_N];   // current X as permutations
    __shared__ short sigc [NPAIR][N_N];   // combo permutations
    __shared__ float affO[NPAIR];
    __shared__ float affC[NPAIR];
    __shared__ float pcv [NPAIR];
    __shared__ int   pcCnt[NPAIR];
    __shared__ int   upt [NPAIR];
    __shared__ float s_norm;

    const int tid  = threadIdx.x;
    const int lane = tid & 31;
    const int wave = tid >> 5;

    // ---- parse dense 0/1 input X into permutation vectors ----
    for (int idx = tid; idx < NPAIR * NN2; idx += NTHREADS) {
        if (Xin[idx] > 0.5f) {
            const int b = idx / NN2;
            const int r = idx - b * NN2;
            sigma[b][r / N_N] = (short)(r % N_N);
        }
    }
    __syncthreads();

    for (int phase = 1; phase <= 2; ++phase) {
        for (int k = 0; k < M_G; ++k) {
            // 1) aff of current X (WMMA gather-reduce)
            wave_aff(K, &sigma[0][0], affO, wave, lane);
            __syncthreads();

            // 2) norm = max off-diagonal aff (single thread: 256 values)
            if (tid == 0) {
                float mx = 0.0f;   // aff*(1-eye): diagonal contributes 0
                for (int b = 0; b < NPAIR; ++b) {
                    const int i = b >> 4, j = b & 15;
                    const float v = (i == j) ? 0.0f : affO[b];
                    mx = fmaxf(mx, v);
                }
                s_norm = mx;
            }
            if (phase == 2) {
                for (int b = tid; b < NPAIR; b += NTHREADS) pcCnt[b] = 0;
            }
            __syncthreads();

            // 3) pair consistency (phase 2): Sum|P-Q| = 2 * #mismatches
            if (phase == 2) {
                for (int idx = tid; idx < M_G * M_G * M_G; idx += NTHREADS) {
                    const int i  = idx >> 8;
                    const int j  = (idx >> 4) & 15;
                    const int kq = idx & 15;
                    const short* sik = sigma[i * 16 + kq];
                    const short* skj = sigma[kq * 16 + j];
                    const short* sij = sigma[i * 16 + j];
                    int cnt = 0;
#pragma unroll
                    for (int a = 0; a < N_N; ++a)
                        cnt += (skj[sik[a]] != sij[a]) ? 1 : 0;
                    atomicAdd(&pcCnt[i * 16 + j], cnt);
                }
            }
            // 4) combo permutations: sigc_{ij} = sig_{kj} o sig_{ik}
            for (int idx = tid; idx < NPAIR * N_N; idx += NTHREADS) {
                const int b = idx / N_N, a = idx - b * N_N;
                const int i = b >> 4, j = b & 15;
                sigc[b][a] = sigma[k * 16 + j][(int)sigma[i * 16 + k][a]];
            }
            __syncthreads();

            if (phase == 2) {
                for (int b = tid; b < NPAIR; b += NTHREADS)
                    pcv[b] = 1.0f - (float)pcCnt[b] / (float)(N_N * M_G);
            }
            // 5) aff of combo (WMMA gather-reduce)
            wave_aff(K, &sigc[0][0], affC, wave, lane);
            __syncthreads();

            // 6) update decision (upper triangle only)
            for (int b = tid; b < NPAIR; b += NTHREADS) {
                const int i = b >> 4, j = b & 15;
                int u = 0;
                if (i < j) {
                    const float ao = affO[b] / s_norm;
                    const float ac = affC[b] / s_norm;
                    if (phase == 1) {
                        u = (ao < ac) ? 1 : 0;
                    } else {
                        const float so = ao * (1.0f - CONST_W) + sqrtf(pcv[b]) * CONST_W;
                        const float sc = ac * (1.0f - CONST_W) +
                                         sqrtf(pcv[i * 16 + k] * pcv[k * 16 + j]) * CONST_W;
                        u = (so < sc) ? 1 : 0;
                    }
                }
                upt[b] = u;
            }
            __syncthreads();

            // 7) apply updates to upper triangle
            for (int idx = tid; idx < NPAIR * N_N; idx += NTHREADS) {
                const int b = idx / N_N, a = idx - b * N_N;
                if (upt[b]) sigma[b][a] = sigc[b][a];
            }
            __syncthreads();

            // 8) sym: lower triangle = inverse of upper
            for (int idx = tid; idx < NPAIR * N_N; idx += NTHREADS) {
                const int b = idx / N_N, a = idx - b * N_N;
                const int i = b >> 4, j = b & 15;
                if (i > j) sigma[b][(int)sigma[j * 16 + i][a]] = (short)a;
            }
            __syncthreads();
        }
    }

    // ---- materialize dense float32 output ----
    for (int idx = tid; idx < NPAIR * NN2; idx += NTHREADS) out[idx] = 0.0f;
    __syncthreads();
    for (int idx = tid; idx < NPAIR * N_N; idx += NTHREADS) {
        const int b = idx / N_N, a = idx - b * N_N;
        out[b * NN2 + a * N_N + (int)sigma[b][a]] = 1.0f;
    }
}

extern "C" void kernel_launch(void* const* d_in, const int* in_sizes, int n_in,
                              void* d_out, int out_size, void* d_ws, size_t ws_size,
                              hipStream_t stream) {
    (void)in_sizes; (void)n_in; (void)d_ws; (void)ws_size; (void)out_size;
    const float* K   = (const float*)d_in[0];   // (16,16,400,400) f32
    const float* Xin = (const float*)d_in[1];   // (16,16,20,20)  f32
    // Single persistent workgroup: the 32 Floyd steps are strictly sequential
    // with intra-step global reductions; total state is ~25 KB of LDS.
    floyd_persistent_kernel<<<1, NTHREADS, 0, stream>>>(K, Xin, (float*)d_out);
}